// FLAMETex_77979426226429
// MI455X (gfx1250) — compile-verified
//
#include <hip/hip_runtime.h>

typedef __attribute__((ext_vector_type(2))) float v2f;
typedef __attribute__((ext_vector_type(8))) float v8f;

#define N_UV        5023
#define BATCH       8
#define KDIM        200
#define TOTAL_ROWS  (3 * N_UV)          // 15069 gathered rows
#define NTILES      ((TOTAL_ROWS + 15) / 16)   // 942
#define LDS_STRIDE  204                 // pad so 16 B-rows hit distinct banks
#define WAVES_PER_BLOCK 8

__global__ __launch_bounds__(32 * WAVES_PER_BLOCK)
void flametex_gather_wmma(const float* __restrict__ texcode,   // [8,200]
                          const float* __restrict__ uv,        // [N_UV,2]
                          const float* __restrict__ mean,      // [786432]
                          const float* __restrict__ basis,     // [786432,200]
                          float* __restrict__ out)             // [8,3,N_UV] flat
{
    __shared__ float lds_tc[16 * LDS_STRIDE];

    const int tid = threadIdx.x;
    // Zero LDS (B-matrix rows 8..15 must multiply as zero), then stage texcode.
    for (int i = tid; i < 16 * LDS_STRIDE; i += blockDim.x) lds_tc[i] = 0.0f;
    __syncthreads();
    for (int i = tid; i < BATCH * KDIM; i += blockDim.x) {
        int b = i / KDIM;
        int k = i - b * KDIM;
        lds_tc[b * LDS_STRIDE + k] = texcode[i];
    }
    __syncthreads();

    const int lane = tid & 31;
    const int wave = tid >> 5;
    const int half = lane >> 4;   // 0: K={0,1}, 1: K={2,3} of each K4 step
    const int m    = lane & 15;   // A: row-in-tile; B/C: column (batch)

    const int tile = blockIdx.x * WAVES_PER_BLOCK + wave;
    if (tile >= NTILES) return;   // wave-uniform: EXEC stays all-1s below

    const int tile_base = tile * 16;

    // Index math for this lane's gathered row r = tile_base + m.
    int r  = tile_base + m;
    int rc = (r < TOTAL_ROWS) ? r : (TOTAL_ROWS - 1);
    int c  = rc / N_UV;                 // channel after ::-1 flip
    int iu = rc - c * N_UV;             // uv point index
    float u  = uv[2 * iu + 0];
    float vv = uv[2 * iu + 1];
    int x = (int)(u * 256.0f);
    x = x < 0 ? 0 : (x > 255 ? 255 : x);
    int y = (int)((1.0f - vv) * 256.0f);
    y = y < 0 ? 0 : (y > 255 ? 255 : y);
    // v = (h*512 + w)*3 + c_orig, h=2y, w=2x, c_orig = 2 - c
    int vidx = ((2 * y) * 512 + 2 * x) * 3 + (2 - c);

    const float* arow = basis + (size_t)vidx * KDIM + 2 * half;
    const int    boff = m * LDS_STRIDE + 2 * half;

    v8f acc = {0.f, 0.f, 0.f, 0.f, 0.f, 0.f, 0.f, 0.f};

    #pragma unroll 10
    for (int k = 0; k < KDIM; k += 4) {
        v2f a = *(const v2f*)(arow + k);                 // global_load_b64 (gather)
        v2f b = *(const v2f*)(&lds_tc[boff + k]);        // ds_load_b64
        // 8 args: (neg_a, A, neg_b, B, c_mod, C, reuse_a, reuse_b)
        acc = __builtin_amdgcn_wmma_f32_16x16x4_f32(
                  false, a, false, b, (short)0, acc, false, false);
    }

    // Mean add: element g of acc is row M = g + 8*half. Lane M holds vidx for
    // that row, so shuffle it over (all lanes active here — before divergence).
    float madd[8];
    #pragma unroll
    for (int g = 0; g < 8; ++g) {
        int vM  = __shfl(vidx, g + 8 * half, 32);
        madd[g] = mean[vM];
    }

    // Store: N = m = batch (only 0..7 valid), row = tile_base + g + 8*half.
    if (m < BATCH) {
        float* outb = out + (size_t)m * TOTAL_ROWS;
        #pragma unroll
        for (int g = 0; g < 8; ++g) {
            int row = tile_base + g + 8 * half;
            if (row < TOTAL_ROWS) outb[row] = acc[g] + madd[g];
        }
    }
}

extern "C" void kernel_launch(void* const* d_in, const int* in_sizes, int n_in,
                              void* d_out, int out_size, void* d_ws, size_t ws_size,
                              hipStream_t stream) {
    const float* texcode = (const float*)d_in[0];   // (8, 200)
    const float* uv      = (const float*)d_in[1];   // (5023, 2)
    const float* mean    = (const float*)d_in[2];   // (1, 1, 786432)
    const float* basis   = (const float*)d_in[3];   // (1, 786432, 200)
    float* out = (float*)d_out;                     // (8, 3, 5023) flat

    const int blocks = (NTILES + WAVES_PER_BLOCK - 1) / WAVES_PER_BLOCK;  // 118
    flametex_gather_wmma<<<blocks, 32 * WAVES_PER_BLOCK, 0, stream>>>(
        texcode, uv, mean, basis, out);
}